// GNNAutoEncoder_6957847020142
// MI455X (gfx1250) — compile-verified
//
#include <hip/hip_runtime.h>
#include <math.h>

typedef __attribute__((ext_vector_type(2))) float v2f;
typedef __attribute__((ext_vector_type(8))) float v8f;

#define NN 256
#define HD 128
#define MLPH 256
#define FOUT 16
#define SUPPS 0.5f
#define EPSL 1e-12f

// ---------------- init: tgtf (also an output), anc=0, Hs init, suppcum=1, lp=0
__global__ __launch_bounds__(256) void k_init(const int* __restrict__ targets,
                                              const float* __restrict__ z,
                                              float* __restrict__ tgtf,
                                              float* __restrict__ anc,
                                              float* __restrict__ Hs,
                                              float* __restrict__ suppcum,
                                              float* __restrict__ lp,
                                              float* __restrict__ out) {
    int b = blockIdx.x;      // row
    int j = threadIdx.x;     // col
    float tv = (j < b) ? (float)targets[b * NN + j] : 0.0f;
    tgtf[b * NN + j] = tv;
    out[b * NN + j]  = tv;                 // first output: tgt [N,N]
    anc[b * NN + j]  = 0.0f;
    if (j < HD) Hs[b * HD + j] = (b == 0) ? z[j] : 0.0f;
    if (b == 0) {
        suppcum[j] = 1.0f;
        if (j == 0) lp[0] = 0.0f;
    }
}

// ---------------- MH = rowmask(d<tlim) * (tgtf @ Hs_in)  ([256x256]@[256x128])
// Row mask applied in the EPILOGUE (zero whole output rows), keeping the inner
// loop pure loads+WMMA. Rows d>=tlim compute garbage-but-finite values (inputs
// are 0/1 adjacency and finite hidden states) and are overwritten with zero.
// Optionally fused: set Hs_in[set_t] = z + pos_emb[set_t] (any concurrent read
// of row set_t only feeds rows that get zeroed in the epilogue).
__global__ __launch_bounds__(256) void k_mh(const float* __restrict__ tgtf,
                                            float* __restrict__ Hs_in,
                                            float* __restrict__ MH,
                                            int tlim, int set_t,
                                            const float* __restrict__ z,
                                            const float* __restrict__ pos_emb) {
    int tid = threadIdx.x;
    if (set_t >= 0 && blockIdx.x == 0 && tid < HD)
        Hs_in[set_t * HD + tid] = z[tid] + pos_emb[set_t * HD + tid];

    int lane = tid & 31, wv = tid >> 5;
    int tile = blockIdx.x * 8 + wv;        // 128 tiles: 16 (M) x 8 (N)
    int mt = tile >> 3, nt = tile & 7;
    int r16 = lane & 15, half = lane >> 4;
    int m = mt * 16 + r16;                 // A row (dst node)
    int n = nt * 16 + r16;                 // B col (hidden dim)

    v8f c = {};
    for (int kb = 0; kb < NN / 4; ++kb) {
        int k = kb * 4 + half * 2;
        v2f a = *(const v2f*)(tgtf + m * NN + k);   // adjacent pair -> b64 load
        v2f b;
        b.x = Hs_in[k * HD + n];
        b.y = Hs_in[(k + 1) * HD + n];
        c = __builtin_amdgcn_wmma_f32_16x16x4_f32(false, a, false, b,
                                                  (short)0, c, false, false);
    }
#pragma unroll
    for (int r = 0; r < 8; ++r) {
        int row = mt * 16 + half * 8 + r;
        MH[row * HD + nt * 16 + r16] = (row < tlim) ? c[r] : 0.0f;
    }
}

// ---------------- Hs_out = relu(Hs_in@Ws + MH@Wn + bias) * (row <= act)
__global__ __launch_bounds__(256) void k_layer(const float* __restrict__ Hs_in,
                                               const float* __restrict__ MH,
                                               const float* __restrict__ Ws,
                                               const float* __restrict__ Wn,
                                               const float* __restrict__ bias,
                                               float* __restrict__ Hs_out,
                                               int act) {
    int tid = threadIdx.x, lane = tid & 31, wv = tid >> 5;
    int tile = blockIdx.x * 8 + wv;
    int mt = tile >> 3, nt = tile & 7;
    int r16 = lane & 15, half = lane >> 4;
    int m = mt * 16 + r16;
    int n = nt * 16 + r16;

    v8f c = {};
    for (int kb = 0; kb < HD / 4; ++kb) {
        int k = kb * 4 + half * 2;
        v2f a1 = *(const v2f*)(Hs_in + m * HD + k);
        v2f b1;
        b1.x = Ws[k * HD + n];
        b1.y = Ws[(k + 1) * HD + n];
        c = __builtin_amdgcn_wmma_f32_16x16x4_f32(false, a1, false, b1,
                                                  (short)0, c, false, false);
        v2f a2 = *(const v2f*)(MH + m * HD + k);
        v2f b2;
        b2.x = Wn[k * HD + n];
        b2.y = Wn[(k + 1) * HD + n];
        c = __builtin_amdgcn_wmma_f32_16x16x4_f32(false, a2, false, b2,
                                                  (short)0, c, false, false);
    }
#pragma unroll
    for (int r = 0; r < 8; ++r) {
        int row = mt * 16 + half * 8 + r;
        int col = nt * 16 + r16;
        float v = c[r] + bias[col];
        v = fmaxf(v, 0.0f);
        if (row > act) v = 0.0f;
        Hs_out[row * HD + col] = v;
    }
}

// ---------------- constvec[n] = bm1[n] + h_t.Wm1[0:128] + z.Wm1[256:384] + pe_t.Wm1[384:512]
__global__ __launch_bounds__(256) void k_constvec(const float* __restrict__ Hs,
                                                  const float* __restrict__ z,
                                                  const float* __restrict__ pos_emb,
                                                  const float* __restrict__ Wm1,
                                                  const float* __restrict__ bm1,
                                                  float* __restrict__ cv, int t) {
    int n = threadIdx.x;
    float s = bm1[n];
    for (int k = 0; k < HD; ++k) {
        s += Hs[t * HD + k]       * Wm1[k * MLPH + n];
        s += z[k]                 * Wm1[(256 + k) * MLPH + n];
        s += pos_emb[t * HD + k]  * Wm1[(384 + k) * MLPH + n];
    }
    cv[n] = s;
}

// ---------------- T = relu(Hs @ Wm1[128:256,:] + constvec)   [256x256], K=128
__global__ __launch_bounds__(256) void k_mlp(const float* __restrict__ Hs,
                                             const float* __restrict__ Wm1,
                                             const float* __restrict__ cv,
                                             float* __restrict__ T) {
    int tid = threadIdx.x, lane = tid & 31, wv = tid >> 5;
    int tile = blockIdx.x * 8 + wv;        // 256 tiles: 16 x 16
    int mt = tile >> 4, nt = tile & 15;
    int r16 = lane & 15, half = lane >> 4;
    int m = mt * 16 + r16;
    int n = nt * 16 + r16;

    v8f c = {};
    for (int kb = 0; kb < HD / 4; ++kb) {
        int k = kb * 4 + half * 2;
        v2f a = *(const v2f*)(Hs + m * HD + k);
        v2f b;
        b.x = Wm1[(128 + k) * MLPH + n];
        b.y = Wm1[(129 + k) * MLPH + n];
        c = __builtin_amdgcn_wmma_f32_16x16x4_f32(false, a, false, b,
                                                  (short)0, c, false, false);
    }
#pragma unroll
    for (int r = 0; r < 8; ++r) {
        int row = mt * 16 + half * 8 + r;
        int col = nt * 16 + r16;
        T[row * MLPH + col] = fmaxf(c[r] + cv[col], 0.0f);
    }
}

// ---------------- tail: logits, sigmoid, suppression, lp accumulate, anc row t
__global__ __launch_bounds__(256) void k_tail(const float* __restrict__ T,
                                              const float* __restrict__ Wm2,
                                              const float* __restrict__ bm2,
                                              const float* __restrict__ tgtf,
                                              float* __restrict__ anc,
                                              float* __restrict__ suppcum,
                                              float* __restrict__ lp, int t) {
    __shared__ float red[256];
    __shared__ float gts[256];
    int i = threadIdx.x;
    gts[i] = tgtf[t * NN + i];
    __syncthreads();

    float logit = bm2[0];
    for (int j = 0; j < MLPH; ++j) logit += T[i * MLPH + j] * Wm2[j];
    float prob = 1.0f / (1.0f + expf(-logit));
    float adj  = prob * suppcum[i];        // read BEFORE this step's update
    float gt   = gts[i];
    float contrib = 0.0f;
    if (i < t)
        contrib = gt * logf(adj + EPSL) + (1.0f - gt) * logf(1.0f - adj + EPSL);
    red[i] = contrib;

    // anc_t[i] = min(1, sum_{p<t} gt[p]*anc[p][i] + gt[i])
    float s = 0.0f;
    for (int p = 1; p < t; ++p) s += gts[p] * anc[p * NN + i];
    s += gt;
    float anct = fminf(1.0f, s);
    anc[t * NN + i] = anct;
    suppcum[i] *= (1.0f - SUPPS * anct);

    __syncthreads();
    for (int off = 128; off > 0; off >>= 1) {
        if (i < off) red[i] += red[i + off];
        __syncthreads();
    }
    if (i == 0) lp[0] += red[0];
}

// ---------------- X_out = Hs @ Wf + bf  [256x16]; also emit lp
__global__ __launch_bounds__(256) void k_xout(const float* __restrict__ Hs,
                                              const float* __restrict__ Wf,
                                              const float* __restrict__ bf,
                                              const float* __restrict__ lp,
                                              float* __restrict__ out) {
    int g = blockIdx.x * 256 + threadIdx.x;   // 4096 outputs
    int row = g >> 4, col = g & 15;
    float s = bf[col];
    for (int k = 0; k < HD; ++k) s += Hs[row * HD + k] * Wf[k * FOUT + col];
    out[NN * NN + g] = s;
    if (g == 0) out[NN * NN + NN * FOUT] = lp[0];
}

extern "C" void kernel_launch(void* const* d_in, const int* in_sizes, int n_in,
                              void* d_out, int out_size, void* d_ws, size_t ws_size,
                              hipStream_t stream) {
    const float* z       = (const float*)d_in[0];
    const int*   targets = (const int*)  d_in[1];
    const float* pos_emb = (const float*)d_in[2];
    const float* Ws1 = (const float*)d_in[3];
    const float* Wn1 = (const float*)d_in[4];
    const float* b1  = (const float*)d_in[5];
    const float* Ws2 = (const float*)d_in[6];
    const float* Wn2 = (const float*)d_in[7];
    const float* b2  = (const float*)d_in[8];
    const float* Wm1 = (const float*)d_in[9];
    const float* bm1 = (const float*)d_in[10];
    const float* Wm2 = (const float*)d_in[11];
    const float* bm2 = (const float*)d_in[12];
    const float* Wf  = (const float*)d_in[13];
    const float* bf  = (const float*)d_in[14];

    float* ws   = (float*)d_ws;
    float* out  = (float*)d_out;
    float* tgtf = ws;                        // 65536
    float* Hs   = tgtf + NN * NN;            // 32768
    float* Hs2  = Hs   + NN * HD;            // 32768
    float* MH   = Hs2  + NN * HD;            // 32768
    float* anc  = MH   + NN * HD;            // 65536
    float* T    = anc  + NN * NN;            // 65536
    float* cv   = T    + NN * MLPH;          // 256
    float* sc   = cv   + MLPH;               // 256
    float* lp   = sc   + NN;                 // 1

    k_init<<<NN, 256, 0, stream>>>(targets, z, tgtf, anc, Hs, sc, lp, out);

    for (int t = 1; t < NN; ++t) {
        // GNN layer 1 (fused set Hs[t] = z + pos_emb[t])
        k_mh<<<16, 256, 0, stream>>>(tgtf, Hs, MH, t, t, z, pos_emb);
        k_layer<<<16, 256, 0, stream>>>(Hs, MH, Ws1, Wn1, b1, Hs2, t);
        // GNN layer 2
        k_mh<<<16, 256, 0, stream>>>(tgtf, Hs2, MH, t, -1, z, pos_emb);
        k_layer<<<16, 256, 0, stream>>>(Hs2, MH, Ws2, Wn2, b2, Hs, t);
        // MLP (K reduced from 512 to 128 via constvec folding)
        k_constvec<<<1, 256, 0, stream>>>(Hs, z, pos_emb, Wm1, bm1, cv, t);
        k_mlp<<<32, 256, 0, stream>>>(Hs, Wm1, cv, T);
        k_tail<<<1, 256, 0, stream>>>(T, Wm2, bm2, tgtf, anc, sc, lp, t);
    }

    // Final full-graph GNN pass (no row masks; all nodes active)
    k_mh<<<16, 256, 0, stream>>>(tgtf, Hs, MH, NN, -1, z, pos_emb);
    k_layer<<<16, 256, 0, stream>>>(Hs, MH, Ws1, Wn1, b1, Hs2, NN - 1);
    k_mh<<<16, 256, 0, stream>>>(tgtf, Hs2, MH, NN, -1, z, pos_emb);
    k_layer<<<16, 256, 0, stream>>>(Hs2, MH, Ws2, Wn2, b2, Hs, NN - 1);
    k_xout<<<16, 256, 0, stream>>>(Hs, Wf, bf, lp, out);
}